// MyLinear_21509196218408
// MI455X (gfx1250) — compile-verified
//
#include <hip/hip_runtime.h>

typedef __attribute__((ext_vector_type(2))) float v2f;
typedef __attribute__((ext_vector_type(8))) float v8f;

#define BATCH_SZ 8192
#define IN_F     4096
#define OUT_F    4096

// Wave tile: 4 (M) x 2 (N) subtiles of 16x16  -> 64 x 32 per wave.
// Workgroup: 8 waves arranged 2 (M) x 4 (N)   -> 128 x 128 per block.

__global__ __launch_bounds__(256, 1)
void fused_early_linear_f32wmma(const float* __restrict__ x,
                                const float* __restrict__ W,
                                const int*   __restrict__ kEarlyPtr,
                                float* __restrict__ out)
{
    const int lane  = threadIdx.x & 31;
    const int wave  = threadIdx.x >> 5;
    const int waveM = wave & 1;   // 0..1
    const int waveN = wave >> 1;  // 0..3

    const int tileM0 = blockIdx.y * 128 + waveM * 64;  // batch-row base of wave
    const int tileN0 = blockIdx.x * 128 + waveN * 32;  // out-col base of wave

    const int half = lane >> 4;   // 0: lanes 0-15 (K=0,1), 1: lanes 16-31 (K=2,3)
    const int l16  = lane & 15;

    const int kEarly = kEarlyPtr[0];   // 1024; multiple of 16 by construction

    // f32 WMMA A-frag: lane (l16, half) loads x[row][k0 + 2*half .. +1]  (b64)
    // f32 WMMA B-frag: same pattern over W rows (B = W^T tile, K contiguous)
    const float* xRow[4];
    const float* wRow[2];
#pragma unroll
    for (int mi = 0; mi < 4; ++mi)
        xRow[mi] = x + (size_t)(tileM0 + mi * 16 + l16) * IN_F + 2 * half;
#pragma unroll
    for (int ni = 0; ni < 2; ++ni)
        wRow[ni] = W + (size_t)(tileN0 + ni * 16 + l16) * IN_F + 2 * half;

    v8f accY[4][2];   // running y accumulation (y1, then y1+y2)
    v8f accS[4][2];   // phase 1: energy sum; after boundary: 0/1 keep-mask
#pragma unroll
    for (int mi = 0; mi < 4; ++mi)
#pragma unroll
        for (int ni = 0; ni < 2; ++ni) { accY[mi][ni] = {}; accS[mi][ni] = {}; }

    // ---------------- Phase 1: k in [0, kEarly) : y1 and cum_squared ----------------
    for (int k0 = 0; k0 < kEarly; k0 += 16) {
#pragma unroll
        for (int kk = 0; kk < 16; kk += 4) {
            v2f a[4], b[2], a2[4], b2[2];
#pragma unroll
            for (int mi = 0; mi < 4; ++mi) {
                a[mi]  = *(const v2f*)(xRow[mi] + k0 + kk);
                a2[mi] = a[mi] * a[mi];
            }
#pragma unroll
            for (int ni = 0; ni < 2; ++ni) {
                b[ni]  = *(const v2f*)(wRow[ni] + k0 + kk);
                b2[ni] = b[ni] * b[ni];
            }
#pragma unroll
            for (int mi = 0; mi < 4; ++mi)
#pragma unroll
                for (int ni = 0; ni < 2; ++ni) {
                    accY[mi][ni] = __builtin_amdgcn_wmma_f32_16x16x4_f32(
                        false, a[mi], false, b[ni], (short)0, accY[mi][ni], false, false);
                    accS[mi][ni] = __builtin_amdgcn_wmma_f32_16x16x4_f32(
                        false, a2[mi], false, b2[ni], (short)0, accS[mi][ni], false, false);
                }
        }
    }

    // ---------------- Confidence test at the k boundary (TAU = 1.0) ----------------
    // passed (-> zero output) when |y1| < sqrt(cum_squared); keep-mask = !passed.
#pragma unroll
    for (int mi = 0; mi < 4; ++mi)
#pragma unroll
        for (int ni = 0; ni < 2; ++ni)
#pragma unroll
            for (int e = 0; e < 8; ++e) {
                float y1 = accY[mi][ni][e];
                float cs = accS[mi][ni][e];
                accS[mi][ni][e] = (__builtin_fabsf(y1) < __builtin_sqrtf(cs)) ? 0.0f : 1.0f;
            }

    // ---------------- Phase 2: k in [kEarly, IN_F) : accumulate y2 ----------------
    for (int k0 = kEarly; k0 < IN_F; k0 += 16) {
#pragma unroll
        for (int kk = 0; kk < 16; kk += 4) {
            v2f a[4], b[2];
#pragma unroll
            for (int mi = 0; mi < 4; ++mi)
                a[mi] = *(const v2f*)(xRow[mi] + k0 + kk);
#pragma unroll
            for (int ni = 0; ni < 2; ++ni)
                b[ni] = *(const v2f*)(wRow[ni] + k0 + kk);
#pragma unroll
            for (int mi = 0; mi < 4; ++mi)
#pragma unroll
                for (int ni = 0; ni < 2; ++ni)
                    accY[mi][ni] = __builtin_amdgcn_wmma_f32_16x16x4_f32(
                        false, a[mi], false, b[ni], (short)0, accY[mi][ni], false, false);
        }
    }

    // ---------------- Store: out = keep ? (y1 + y2) : 0 ----------------
    // C/D layout: VGPR e -> row (e + 8*half), col l16 within each 16x16 subtile.
#pragma unroll
    for (int mi = 0; mi < 4; ++mi)
#pragma unroll
        for (int ni = 0; ni < 2; ++ni)
#pragma unroll
            for (int e = 0; e < 8; ++e) {
                size_t row = (size_t)(tileM0 + mi * 16 + e + 8 * half);
                size_t col = (size_t)(tileN0 + ni * 16 + l16);
                out[row * OUT_F + col] = accS[mi][ni][e] * accY[mi][ni][e];
            }
}

extern "C" void kernel_launch(void* const* d_in, const int* in_sizes, int n_in,
                              void* d_out, int out_size, void* d_ws, size_t ws_size,
                              hipStream_t stream) {
    (void)in_sizes; (void)n_in; (void)out_size; (void)d_ws; (void)ws_size;
    const float* x    = (const float*)d_in[0];
    const float* W    = (const float*)d_in[1];
    const int*   kptr = (const int*)d_in[2];
    float*       out  = (float*)d_out;

    dim3 grid(OUT_F / 128, BATCH_SZ / 128);  // 32 x 64 blocks
    dim3 block(256, 1, 1);                   // 8 waves (wave32)
    fused_early_linear_f32wmma<<<grid, block, 0, stream>>>(x, W, kptr, out);
}